// AttentionAlignmentLoss_58050777972822
// MI455X (gfx1250) — compile-verified
//
#include <hip/hip_runtime.h>
#include <hip/hip_bf16.h>

// Problem constants (from the reference)
#define B_  16
#define T_  512
#define F_  3000
#define CHUNKS (F_ / 4)          // 750 K-chunks of 4 for WMMA f32 16x16x4
#define WAVES  16                // waves per block
#define BLOCK  (WAVES * 32)      // 512 threads, wave32
#define TILES  (B_ * T_ / 16)    // 512 blocks, one 16-token tile each
#define INTERP 4
#define EPS_   1e-8f

typedef __attribute__((ext_vector_type(2))) float v2f;
typedef __attribute__((ext_vector_type(8))) float v8f;

#define WMMA_F32(acc, a) \
  __builtin_amdgcn_wmma_f32_16x16x4_f32(false, (a), false, (a), (short)0, (acc), false, false)

// Stage 1: per 16-token tile, compute per-row ||pred||^2 via WMMA Gram diagonal,
// then the analytic-window dot(pred, gt) and ||gt||^2, emit per-row loss.
__global__ __launch_bounds__(BLOCK) void attn_loss_stage1(
    const float* __restrict__ pred,      // [B,T,F]
    const float* __restrict__ ts,        // [B,T,2]
    const float* __restrict__ mask,      // [B,T]
    float* __restrict__ row_loss)        // [B*T]
{
  __shared__ float sdiag[WAVES][16];

  const int tid  = threadIdx.x;
  const int wave = tid >> 5;
  const int lane = tid & 31;
  const int blk  = blockIdx.x;
  const int b    = blk >> 5;            // 32 tiles per batch row
  const int t0   = (blk & 31) << 4;

  // A-fragment mapping for V_WMMA_F32_16X16X4_F32:
  //   lane<16 : VGPR0=X[M=lane,K=0], VGPR1=X[M=lane,K=1]
  //   lane>=16: VGPR0=X[M=lane-16,K=2], VGPR1=X[M=lane-16,K=3]
  const int rowInTile = lane & 15;
  const int kHalf     = lane >> 4;
  const float* rowPtr = pred + (size_t)(b * T_ + t0 + rowInTile) * F_;

  // Wave w owns chunk slots i -> chunk (w + i*16); per-slot address stride is
  // 16 chunks * 4 floats = 64 floats. Slot count: 47 for waves 0..13, else 46.
  const float* p0 = rowPtr + (wave << 2) + (kHalf << 1);
  const int nIter = (CHUNKS - wave + WAVES - 1) / WAVES;   // wave-uniform 46/47

  v8f acc0 = {};
  v8f acc1 = {};

  // Software-pipelined: 'a' always holds the fragment for the slot about to be
  // consumed; the next slot's load issues before the current WMMA retires.
  v2f a = *(const v2f*)p0;
  for (int i = 0; i < 46; i += 2) {
    v2f bfrag = *(const v2f*)(p0 + (size_t)(i + 1) * 64);  // slot i+1, always valid
    __builtin_prefetch(p0 + (size_t)(i + 1) * 64 + 1024, 0, 1);  // speculative GL2
    acc0 = WMMA_F32(acc0, a);                              // consume slot i
    // Preload slot i+2; clamp to slot 0 (dummy) when past this wave's range so
    // the last tile row never reads past the end of the tensor.
    size_t noff = (i + 2 < nIter) ? (size_t)(i + 2) * 64 : 0;
    v2f cfrag = *(const v2f*)(p0 + noff);
    acc1 = WMMA_F32(acc1, bfrag);                          // consume slot i+1
    a = cfrag;
  }
  if (nIter == 47) acc0 = WMMA_F32(acc0, a);               // tail slot 46 (waves 0..13)

  v8f acc = acc0 + acc1;

  // Extract Gram diagonal: D[i][i] sits at (VGPR i, lane i) for i<8 and
  // (VGPR i-8, lane i+16) for i>=8.
  float carr[8];
#pragma unroll
  for (int i = 0; i < 8; ++i) carr[i] = acc[i];
  if (lane < 8)        sdiag[wave][lane]      = carr[lane];
  else if (lane >= 24) sdiag[wave][lane - 16] = carr[lane - 24];
  __syncthreads();

  if (tid < 16) {
    float psq = 0.f;
#pragma unroll
    for (int w = 0; w < WAVES; ++w) psq += sdiag[w][tid];

    const int row = t0 + tid;
    const size_t gi = (size_t)(b * T_ + row);
    const float s_ts = ts[gi * 2 + 0];
    const float e_ts = ts[gi * 2 + 1];
    // int() truncation toward zero (timestamps nonnegative), then clips —
    // exactly the reference semantics.
    int sf = (int)(s_ts * 12.5f);
    int ef = (int)(e_ts * 12.5f);
    sf = min(max(sf, 0), F_ - 1);
    ef = max(sf + 1, min(ef + 1, F_));

    const int lo = max(sf - INTERP, 0);
    const int hi = min(ef + INTERP, F_);
    const float* p = pred + gi * F_;
    float dot = 0.f, gsq = 0.f;
    for (int f = lo; f < hi; ++f) {
      float w;
      if (f < sf)      w = (float)(f - sf + INTERP + 1) / 5.0f;  // up-ramp
      else if (f < ef) w = 1.0f;                                  // core
      else             w = 1.0f - (float)(f - ef + 1) / 5.0f;     // down-ramp
      dot = fmaf(p[f], w, dot);
      gsq = fmaf(w, w, gsq);
    }
    const float pn = fmaxf(sqrtf(psq), EPS_);
    const float gn = fmaxf(sqrtf(gsq), EPS_);
    const float cosv = dot / (pn * gn);
    row_loss[gi] = (1.0f - cosv) * mask[gi];
  }
}

// Stage 2: deterministic tree reduction of 8192 per-row losses + mask sum.
__global__ __launch_bounds__(1024) void attn_loss_stage2(
    const float* __restrict__ row_loss,
    const float* __restrict__ mask,
    float* __restrict__ out, int n)
{
  __shared__ float sl[1024];
  __shared__ float sm[1024];
  const int tid = threadIdx.x;
  float l = 0.f, m = 0.f;
  for (int i = tid; i < n; i += 1024) {
    l += row_loss[i];
    m += mask[i];
  }
  sl[tid] = l;
  sm[tid] = m;
  __syncthreads();
  for (int s = 512; s > 0; s >>= 1) {
    if (tid < s) {
      sl[tid] += sl[tid + s];
      sm[tid] += sm[tid + s];
    }
    __syncthreads();
  }
  if (tid == 0) out[0] = sl[0] / fmaxf(sm[0], 1.0f);
}

extern "C" void kernel_launch(void* const* d_in, const int* in_sizes, int n_in,
                              void* d_out, int out_size, void* d_ws, size_t ws_size,
                              hipStream_t stream) {
  const float* pred = (const float*)d_in[0];   // predicted_attn [B,T,F] f32
  const float* ts   = (const float*)d_in[1];   // token_timestamps [B,T,2] f32
  const float* mask = (const float*)d_in[2];   // attention_mask [B,T] f32
  float* row_loss   = (float*)d_ws;            // B*T floats = 32 KB scratch

  attn_loss_stage1<<<TILES, BLOCK, 0, stream>>>(pred, ts, mask, row_loss);
  attn_loss_stage2<<<1, 1024, 0, stream>>>(row_loss, mask, (float*)d_out, B_ * T_);
}